// ROIPooling_47399259078942
// MI455X (gfx1250) — compile-verified
//
#include <hip/hip_runtime.h>
#include <cstddef>

typedef float v4f __attribute__((ext_vector_type(4)));

#define OUT_H 7
#define OUT_W 7
#define H_IN  57
#define W_IN  57
#define C4    128   // 512 channels / 4 floats per b128 load

// Adaptive max pool, NHWC (128,57,57,512) -> (128,7,7,512).
// Bins: [8,8,8,8,8,8,9] in both H and W (offsets 0,8,...,48,57).
//
// Grid carries the output-position decomposition so ALL bin logic is scalar:
//   blockIdx.x = ow (0..6), blockIdx.y = oh (0..6), blockIdx.z = n (0..127)
//   threadIdx.x = c4 (0..127), one float4 of channels per thread.
// A wave32 issues 32 consecutive b128 loads = 512 contiguous bytes -> fully
// coalesced. Input is 852 MB read exactly once (bins disjoint), far larger
// than the 192 MB L2, so loads are non-temporal.
__global__ __launch_bounds__(128) void adaptive_maxpool_nhwc_47399259078942(
    const v4f* __restrict__ x, v4f* __restrict__ out)
{
    const int c4 = threadIdx.x;
    const int ow = blockIdx.x;
    const int oh = blockIdx.y;
    const int n  = blockIdx.z;

    const int h0       = oh * 8;
    const int rows     = (oh == OUT_H - 1) ? 9 : 8;   // scalar
    const int w0       = ow * 8;
    const bool lastcol = (ow == OUT_W - 1);           // scalar

    const v4f* p = x + ((size_t)((n * H_IN + h0) * W_IN + w0)) * C4 + c4;

    v4f m = { -3.402823466e+38f, -3.402823466e+38f,
              -3.402823466e+38f, -3.402823466e+38f };

    for (int r = 0; r < rows; ++r) {
        // 8 unconditional, independent global_load_b128 (NT) in flight,
        // then the v_max_num_f32 chain consumes them.
        #pragma unroll
        for (int w = 0; w < 8; ++w) {
            v4f v = __builtin_nontemporal_load(p + (size_t)w * C4);
            m.x = __builtin_fmaxf(m.x, v.x);
            m.y = __builtin_fmaxf(m.y, v.y);
            m.z = __builtin_fmaxf(m.z, v.z);
            m.w = __builtin_fmaxf(m.w, v.w);
        }
        if (lastcol) {   // scalar-uniform branch, 9th column of last W bin
            v4f v = __builtin_nontemporal_load(p + (size_t)8 * C4);
            m.x = __builtin_fmaxf(m.x, v.x);
            m.y = __builtin_fmaxf(m.y, v.y);
            m.z = __builtin_fmaxf(m.z, v.z);
            m.w = __builtin_fmaxf(m.w, v.w);
        }
        p += (size_t)W_IN * C4;   // next input row
    }

    out[((size_t)((n * OUT_H + oh) * OUT_W + ow)) * C4 + c4] = m;
}

extern "C" void kernel_launch(void* const* d_in, const int* in_sizes, int n_in,
                              void* d_out, int out_size, void* d_ws, size_t ws_size,
                              hipStream_t stream) {
    (void)in_sizes; (void)n_in; (void)d_ws; (void)ws_size; (void)out_size;

    const v4f* x = (const v4f*)d_in[0];   // float32 input viewed as float4
    v4f* out     = (v4f*)d_out;           // float32 output viewed as float4

    dim3 grid(OUT_W, OUT_H, 128);         // (ow, oh, n) -> 6272 blocks
    dim3 block(C4, 1, 1);                 // c4 -> 4 waves of 32

    adaptive_maxpool_nhwc_47399259078942<<<grid, block, 0, stream>>>(x, out);
}